// ModuleAttentionPropHoi_33861522162536
// MI455X (gfx1250) — compile-verified
//
#include <hip/hip_runtime.h>
#include <hip/hip_bf16.h>

typedef float v2f __attribute__((ext_vector_type(2)));
typedef float v8f __attribute__((ext_vector_type(8)));

#define B_ 2
#define N_ 1024
#define M_ 2048
#define D_ 512
#define H_ 64
#define NEG_BIAS_ 1e23f

// ---------------------------------------------------------------------------
// Generic row-major f32 GEMM using V_WMMA_F32_16X16X4_F32.
// One wave computes one 16x16 tile of C = A(MxK) @ B(KxN), batched via strides.
// Launch with exactly (batch*tilesPerBatch) waves; 8 waves per 256-thread block.
// ---------------------------------------------------------------------------
__global__ void wmma_gemm_f32(const float* __restrict__ A, long sA,
                              const float* __restrict__ Bm, long sB,
                              float* __restrict__ C, long sC,
                              int K, int Nn, int tilesPerBatch, int tilesN)
{
    int wave = (int)((blockIdx.x * blockDim.x + threadIdx.x) >> 5);
    int lane = threadIdx.x & 31;
    int batch = wave / tilesPerBatch;
    int t     = wave % tilesPerBatch;
    int tm = t / tilesN, tn = t % tilesN;

    const float* Ab = A  + (long)batch * sA;
    const float* Bb = Bm + (long)batch * sB;
    float*       Cb = C  + (long)batch * sC;

    int mn = lane & 15;            // M index (A) / N index (B,C) within tile
    int kb = (lane >> 4) << 1;     // K sub-offset {0,2} by lane half
    int i0 = tm * 16, j0 = tn * 16;

    const float* arow = Ab + (long)(i0 + mn) * K + kb;          // A[i0+m][kb + k]
    const float* bcol = Bb + (long)kb * Nn + (j0 + mn);         // B[kb + k][j0+n]

    v8f acc = {};
    for (int k = 0; k < K; k += 4) {
        v2f a; a.x = arow[k];                a.y = arow[k + 1];
        v2f b; b.x = bcol[(long)k * Nn];     b.y = bcol[(long)k * Nn + Nn];
        acc = __builtin_amdgcn_wmma_f32_16x16x4_f32(
                  false, a, false, b, (short)0, acc, false, false);
    }

    int half = (lane >> 4) << 3;   // row offset {0,8} by lane half
#pragma unroll
    for (int r = 0; r < 8; ++r) {
        Cb[(long)(i0 + r + half) * Nn + (j0 + mn)] = acc[r];
    }
}

// ---------------------------------------------------------------------------
// Gate GEMM: acc = [upd, ctxt](2048 x 1024) @ W_gate(1024 x 512), fused
// epilogue: g = sigmoid(acc + b_gate); out = g*upd + (1-g)*ctxt.
// ---------------------------------------------------------------------------
__global__ void gate_wmma_kernel(const float* __restrict__ upd,
                                 const float* __restrict__ ctxt,
                                 const float* __restrict__ Wg,
                                 const float* __restrict__ bg,
                                 float* __restrict__ out)
{
    const int K = 2 * D_, TN = D_ / 16;
    int wave = (int)((blockIdx.x * blockDim.x + threadIdx.x) >> 5);
    int lane = threadIdx.x & 31;
    int tm = wave / TN, tn = wave % TN;

    int mn = lane & 15;
    int kb = (lane >> 4) << 1;
    int i0 = tm * 16, j0 = tn * 16;

    const float* arowU = upd  + (long)(i0 + mn) * D_;
    const float* arowC = ctxt + (long)(i0 + mn) * D_ - D_;   // shifted for k>=D_
    const float* bcol  = Wg + (long)kb * D_ + (j0 + mn);

    v8f acc = {};
    for (int k = 0; k < K; k += 4) {
        int ka = k + kb;                       // all lanes agree on side (D_%4==0)
        const float* src = (ka < D_) ? arowU : arowC;
        v2f a; a.x = src[ka];                a.y = src[ka + 1];
        v2f b; b.x = bcol[(long)k * D_];     b.y = bcol[(long)k * D_ + D_];
        acc = __builtin_amdgcn_wmma_f32_16x16x4_f32(
                  false, a, false, b, (short)0, acc, false, false);
    }

    int half = (lane >> 4) << 3;
#pragma unroll
    for (int r = 0; r < 8; ++r) {
        int row = i0 + r + half;
        int col = j0 + mn;
        long off = (long)row * D_ + col;
        float u = upd[off], c = ctxt[off];
        float g = 1.0f / (1.0f + __expf(-(acc[r] + bg[col])));
        out[off] = g * u + (1.0f - g) * c;
    }
}

// ---------------------------------------------------------------------------
// scores[b,i,j] = v_out . relu(left[b,i] + right[b,j] + dist_emb[bucket] + b_h)
//                 + b_out - (1-mask)*NEG_BIAS
// grid = (N/256, N, B), block = 256; one block handles a fixed (b,i).
// ---------------------------------------------------------------------------
__global__ void scores_kernel(const float* __restrict__ left,
                              const float* __restrict__ right,
                              const float* __restrict__ dist_emb,
                              const float* __restrict__ b_h,
                              const float* __restrict__ v_out,
                              const float* __restrict__ b_out,
                              const float* __restrict__ mask,
                              const int*   __restrict__ span_begin,
                              float* __restrict__ scores)
{
    __shared__ float sL[H_], sBh[H_], sV[H_], sDist[10 * H_];
    int b = blockIdx.z, i = blockIdx.y;
    int j = blockIdx.x * 256 + threadIdx.x;
    int tid = threadIdx.x;

    if (tid < H_) {
        sL[tid]  = left[((long)b * N_ + i) * H_ + tid];
        sBh[tid] = b_h[tid];
        sV[tid]  = v_out[tid];
    }
    for (int t = tid; t < 10 * H_; t += 256) sDist[t] = dist_emb[t];
    __syncthreads();

    int sbi = span_begin[b * N_ + i];
    int sbj = span_begin[b * N_ + j];
    int d = abs(sbi - sbj);
    int bk = (d <= 4) ? d : (d <= 7) ? 5 : (d <= 15) ? 6 : (d <= 31) ? 7
                                                        : (d <= 63) ? 8 : 9;
    const float* R  = right + ((long)b * N_ + j) * H_;
    const float* Dk = sDist + bk * H_;

    float s = 0.0f;
#pragma unroll 8
    for (int h = 0; h < H_; ++h) {
        float t = sL[h] + R[h] + Dk[h] + sBh[h];
        s += sV[h] * fmaxf(t, 0.0f);
    }
    long off = ((long)b * N_ + i) * N_ + j;
    scores[off] = s + b_out[0] - (1.0f - mask[off]) * NEG_BIAS_;
}

// ---------------------------------------------------------------------------
// In-place row softmax over last axis (N=1024). grid=(N,B), block=256, 4/thread.
// ---------------------------------------------------------------------------
__global__ void softmax_kernel(float* __restrict__ scores)
{
    __shared__ float red[256];
    long base = ((long)blockIdx.y * N_ + blockIdx.x) * N_;
    int tid = threadIdx.x;

    float v[4];
    float mx = -INFINITY;
#pragma unroll
    for (int q = 0; q < 4; ++q) {
        v[q] = scores[base + tid + q * 256];
        mx = fmaxf(mx, v[q]);
    }
    red[tid] = mx; __syncthreads();
    for (int s = 128; s > 0; s >>= 1) {
        if (tid < s) red[tid] = fmaxf(red[tid], red[tid + s]);
        __syncthreads();
    }
    float rowmax = red[0]; __syncthreads();

    float sum = 0.0f;
#pragma unroll
    for (int q = 0; q < 4; ++q) { v[q] = __expf(v[q] - rowmax); sum += v[q]; }
    red[tid] = sum; __syncthreads();
    for (int s = 128; s > 0; s >>= 1) {
        if (tid < s) red[tid] += red[tid + s];
        __syncthreads();
    }
    float inv = 1.0f / red[0];
#pragma unroll
    for (int q = 0; q < 4; ++q) scores[base + tid + q * 256] = v[q] * inv;
}

// ---------------------------------------------------------------------------
// cand_out[b, idx[b,i], :] = upd[b, i, :] for valid i. grid=(D/256, N, B).
// Indices are distinct within a batch -> no write conflicts.
// ---------------------------------------------------------------------------
__global__ void scatter_kernel(const float* __restrict__ upd,
                               const int* __restrict__ idx,
                               const int* __restrict__ span_len,
                               float* __restrict__ cand_out)
{
    int b = blockIdx.z, i = blockIdx.y;
    int d = blockIdx.x * 256 + threadIdx.x;
    if (i < span_len[b]) {
        int row = idx[b * N_ + i];
        cand_out[((long)b * M_ + row) * D_ + d] = upd[((long)b * N_ + i) * D_ + d];
    }
}

extern "C" void kernel_launch(void* const* d_in, const int* in_sizes, int n_in,
                              void* d_out, int out_size, void* d_ws, size_t ws_size,
                              hipStream_t stream)
{
    const float* update      = (const float*)d_in[0];
    const float* square_mask = (const float*)d_in[1];
    const float* cand        = (const float*)d_in[2];
    const int*   span_begin  = (const int*)  d_in[3];
    /* d_in[4] = span_end (unused by reference) */
    const int*   prune_idx   = (const int*)  d_in[5];
    const int*   span_len    = (const int*)  d_in[6];
    const float* W_left      = (const float*)d_in[7];
    const float* W_right     = (const float*)d_in[8];
    const float* b_h         = (const float*)d_in[9];
    const float* dist_emb    = (const float*)d_in[10];
    const float* v_out       = (const float*)d_in[11];
    const float* b_out       = (const float*)d_in[12];
    const float* W_gate      = (const float*)d_in[13];
    const float* b_gate      = (const float*)d_in[14];

    const long BND = (long)B_ * N_ * D_;   // 1,048,576
    const long BNH = (long)B_ * N_ * H_;   // 131,072
    const long BNN = (long)B_ * N_ * N_;   // 2,097,152
    const long BMD = (long)B_ * M_ * D_;   // 2,097,152

    float* ws     = (float*)d_ws;
    float* upd0   = ws;
    float* upd1   = upd0 + BND;
    float* left   = upd1 + BND;
    float* right  = left + BNH;
    float* scores = right + BNH;
    float* ctxt   = scores + BNN;

    hipMemcpyAsync(upd0, update, BND * sizeof(float),
                   hipMemcpyDeviceToDevice, stream);

    float* cur = upd0;
    float* nxt = upd1;
    for (int it = 0; it < 2; ++it) {
        // left/right projections: (2048x512)@(512x64); 128x4 = 512 tiles -> 64 blocks
        wmma_gemm_f32<<<64, 256, 0, stream>>>(cur, 0, W_left, 0, left, 0,
                                              D_, H_, 512, H_ / 16);
        wmma_gemm_f32<<<64, 256, 0, stream>>>(cur, 0, W_right, 0, right, 0,
                                              D_, H_, 512, H_ / 16);

        scores_kernel<<<dim3(N_ / 256, N_, B_), 256, 0, stream>>>(
            left, right, dist_emb, b_h, v_out, b_out, square_mask,
            span_begin, scores);

        softmax_kernel<<<dim3(N_, B_), 256, 0, stream>>>(scores);

        // ctxt[b] = probs[b](1024x1024) @ cur[b](1024x512); 64x32=2048 tiles/batch
        wmma_gemm_f32<<<512, 256, 0, stream>>>(scores, (long)N_ * N_,
                                               cur, (long)N_ * D_,
                                               ctxt, (long)N_ * D_,
                                               N_, D_, 2048, D_ / 16);

        // gated update -> nxt; 128x32 = 4096 tiles -> 512 blocks
        gate_wmma_kernel<<<512, 256, 0, stream>>>(cur, ctxt, W_gate, b_gate, nxt);

        float* tmp = cur; cur = nxt; nxt = tmp;
    }

    float* outCand = (float*)d_out;
    float* outUpd  = outCand + BMD;
    hipMemcpyAsync(outCand, cand, BMD * sizeof(float),
                   hipMemcpyDeviceToDevice, stream);
    hipMemcpyAsync(outUpd, cur, BND * sizeof(float),
                   hipMemcpyDeviceToDevice, stream);
    scatter_kernel<<<dim3(D_ / 256, N_, B_), 256, 0, stream>>>(
        cur, prune_idx, span_len, outCand);
}